// RegularizedFMNet_4501125726486
// MI455X (gfx1250) — compile-verified
//
#include <hip/hip_runtime.h>

typedef __attribute__((ext_vector_type(2))) float v2f;
typedef __attribute__((ext_vector_type(8))) float v8f;

#define LMBDA 100.0f
#define B_ 8
#define V_ 8192
#define C_ 256
#define K_ 128
#define PITCH 129
#define KT 16          // V-tile depth staged in LDS
#define EPITCH 20      // Et tile row pitch (floats): 80B rows -> 16B aligned, bank-conflict-free

// ---------------------------------------------------------------------------
// Kernel 1: per-batch projection GEMM   Out[b] = Et[b] (K x V) * F[b] (V x C)
// Block: 256 threads (8 waves). Each wave computes a 16(M) x 64(N) strip.
// Et/F tiles are staged into LDS with CDNA5 async global->LDS b128 copies
// (ASYNCcnt), double-buffered so the next tile's DMA overlaps this tile's
// WMMAs. grid = (C/64, B)
// ---------------------------------------------------------------------------
__global__ __launch_bounds__(256) void proj_gemm(const float* __restrict__ Et,
                                                 const float* __restrict__ F,
                                                 float* __restrict__ Out)
{
    __shared__ float etTile[2][K_ * EPITCH];   // 2 x 10 KB
    __shared__ float fTile[2][KT * 64];        // 2 x 4 KB

    const int strip = blockIdx.x;              // N strip of 64
    const int b     = blockIdx.y;
    const int tid   = threadIdx.x;
    const int wave  = tid >> 5;
    const int lane  = tid & 31;
    const int m     = lane & 15;
    const int kg    = lane >> 4;               // 0 or 1 (K half-group)

    const float* etb = Et + (size_t)b * K_ * V_;
    const float* fb  = F  + (size_t)b * V_ * C_ + strip * 64;
    float*       o   = Out + (size_t)b * K_ * C_ + (size_t)(wave * 16) * C_ + strip * 64;

    // staging assignments (b128 = 4 floats per lane-transfer)
    const int erow = tid >> 2;                 // Et rows 0..63 (+64 on 2nd issue)
    const int ecg  = tid & 3;                  // Et col group (4 floats), KT=16
    const int frow = tid >> 4;                 // F rows 0..15
    const int fcg  = tid & 15;                 // F col group (4 floats), 64 cols

    auto issue_tile = [&](int v0, int buf) {
        {   // Et tile: 128 rows x KT cols, two b128 transfers per thread
            unsigned l0 = (unsigned)(uintptr_t)&etTile[buf][erow * EPITCH + ecg * 4];
            const float* g0 = etb + (size_t)erow * V_ + v0 + ecg * 4;
            asm volatile("global_load_async_to_lds_b128 %0, %1, off"
                         :: "v"(l0), "v"(g0) : "memory");
            unsigned l1 = (unsigned)(uintptr_t)&etTile[buf][(erow + 64) * EPITCH + ecg * 4];
            const float* g1 = etb + (size_t)(erow + 64) * V_ + v0 + ecg * 4;
            asm volatile("global_load_async_to_lds_b128 %0, %1, off"
                         :: "v"(l1), "v"(g1) : "memory");
        }
        {   // F tile: KT rows x 64 cols, one b128 transfer per thread
            unsigned l2 = (unsigned)(uintptr_t)&fTile[buf][frow * 64 + fcg * 4];
            const float* g2 = fb + (size_t)(v0 + frow) * C_ + fcg * 4;
            asm volatile("global_load_async_to_lds_b128 %0, %1, off"
                         :: "v"(l2), "v"(g2) : "memory");
        }
    };

    v8f acc0 = {}, acc1 = {}, acc2 = {}, acc3 = {};

    issue_tile(0, 0);
    int buf = 0;
    for (int v0 = 0; v0 < V_; v0 += KT) {
        // wait for my async copies, then make them visible workgroup-wide
        asm volatile("s_wait_asynccnt 0x0" ::: "memory");
        __syncthreads();
        if (v0 + KT < V_) issue_tile(v0 + KT, buf ^ 1);

#pragma unroll
        for (int k = 0; k < KT; k += 4) {
            // A fragment: row (wave*16+m), K = k + 2*kg + {0,1} -> one ds b64
            v2f a = ((const v2f*)(&etTile[buf][(wave * 16 + m) * EPITCH + k]))[kg];
            // B fragments: VGPR j holds row K = k + 2*kg + j, col = n0 + (lane&15)
            const float* fr0 = &fTile[buf][(k + 2 * kg) * 64 + m];
            const float* fr1 = fr0 + 64;
            v2f b0 = { fr0[0],  fr1[0]  };
            v2f b1 = { fr0[16], fr1[16] };
            v2f b2 = { fr0[32], fr1[32] };
            v2f b3 = { fr0[48], fr1[48] };
            acc0 = __builtin_amdgcn_wmma_f32_16x16x4_f32(false, a, false, b0, (short)0, acc0, false, false);
            acc1 = __builtin_amdgcn_wmma_f32_16x16x4_f32(false, a, false, b1, (short)0, acc1, false, false);
            acc2 = __builtin_amdgcn_wmma_f32_16x16x4_f32(false, a, false, b2, (short)0, acc2, false, false);
            acc3 = __builtin_amdgcn_wmma_f32_16x16x4_f32(false, a, false, b3, (short)0, acc3, false, false);
        }
        __syncthreads();
        buf ^= 1;
    }

    // C/D layout: VGPR i -> (M = i + kg*8, N = lane&15)
#pragma unroll
    for (int i = 0; i < 8; ++i) {
        const int row = i + kg * 8;
        float* p = o + (size_t)row * C_ + m;
        p[0]  = acc0[i];
        p[16] = acc1[i];
        p[32] = acc2[i];
        p[48] = acc3[i];
    }
}

// ---------------------------------------------------------------------------
// Kernel 2: gram GEMM   Out[b] = X[b] (K x C) * A[b]^T (C x K)
// grid = (K/64, B) ; inputs are small (resident in L2), plain loads are fine
// ---------------------------------------------------------------------------
__global__ __launch_bounds__(256) void gram_gemm(const float* __restrict__ X,
                                                 const float* __restrict__ A,
                                                 float* __restrict__ Out)
{
    const int strip = blockIdx.x;          // N strip of 64 over K_
    const int b     = blockIdx.y;
    const int wave  = threadIdx.x >> 5;
    const int lane  = threadIdx.x & 31;
    const int m     = lane & 15;
    const int kg    = lane >> 4;

    const float* x  = X   + (size_t)b * K_ * C_ + (size_t)(wave * 16) * C_;
    const float* ar = A   + (size_t)b * K_ * C_ + (size_t)(strip * 64) * C_;
    float*       o  = Out + (size_t)b * K_ * K_ + (size_t)(wave * 16) * K_ + strip * 64;

    v8f acc0 = {}, acc1 = {}, acc2 = {}, acc3 = {};

    for (int c = 0; c < C_; c += 4) {
        v2f a = ((const v2f*)(x + (size_t)m * C_ + c))[kg];
        // B[K=2kg+j, N=n] = A[n, c+2kg+j] -> contiguous pair per lane (b64 load)
        v2f b0 = ((const v2f*)(ar + (size_t)(0 * 16 + m) * C_ + c))[kg];
        v2f b1 = ((const v2f*)(ar + (size_t)(1 * 16 + m) * C_ + c))[kg];
        v2f b2 = ((const v2f*)(ar + (size_t)(2 * 16 + m) * C_ + c))[kg];
        v2f b3 = ((const v2f*)(ar + (size_t)(3 * 16 + m) * C_ + c))[kg];
        acc0 = __builtin_amdgcn_wmma_f32_16x16x4_f32(false, a, false, b0, (short)0, acc0, false, false);
        acc1 = __builtin_amdgcn_wmma_f32_16x16x4_f32(false, a, false, b1, (short)0, acc1, false, false);
        acc2 = __builtin_amdgcn_wmma_f32_16x16x4_f32(false, a, false, b2, (short)0, acc2, false, false);
        acc3 = __builtin_amdgcn_wmma_f32_16x16x4_f32(false, a, false, b3, (short)0, acc3, false, false);
    }

#pragma unroll
    for (int i = 0; i < 8; ++i) {
        const int row = i + kg * 8;
        float* p = o + (size_t)row * K_ + m;
        p[0]  = acc0[i];
        p[16] = acc1[i];
        p[32] = acc2[i];
        p[48] = acc3[i];
    }
}

// ---------------------------------------------------------------------------
// Kernel 3: resolvent mask  D[b,i,j]  (gamma = 0.5 -> sqrt)
// grid = (B), block = 128
// ---------------------------------------------------------------------------
__global__ __launch_bounds__(128) void mask_kernel(const float* __restrict__ ex,
                                                   const float* __restrict__ ey,
                                                   float* __restrict__ D)
{
    __shared__ float sx[K_];
    __shared__ float sy[K_];
    __shared__ float red[K_];
    const int b = blockIdx.x;
    const int t = threadIdx.x;

    sx[t] = ex[b * K_ + t];
    sy[t] = ey[b * K_ + t];
    red[t] = fmaxf(sx[t], sy[t]);
    __syncthreads();
    for (int s = 64; s > 0; s >>= 1) {
        if (t < s) red[t] = fmaxf(red[t], red[t + s]);
        __syncthreads();
    }
    const float scale = red[0];

    const float g2 = sqrtf(sy[t] / scale);
    const float q2 = g2 * g2 + 1.0f;
    const float re2 = g2 / q2;
    const float im2 = 1.0f / q2;
    float* drow = D + (size_t)b * K_ * K_ + (size_t)t * K_;
    for (int j = 0; j < K_; ++j) {
        const float g1 = sqrtf(sx[j] / scale);
        const float q1 = g1 * g1 + 1.0f;
        const float re = re2 - g1 / q1;
        const float im = im2 - 1.0f / q1;
        drow[j] = re * re + im * im;
    }
}

// ---------------------------------------------------------------------------
// Kernel 4: SPD solve  (AAt[b] + LMBDA*diag(D[b,i,:])) x = BAt[b,i,:]
// one block (128 threads, thread r owns row r) per (b,i); M in LDS, pitch 129
// ---------------------------------------------------------------------------
__global__ __launch_bounds__(128) void solve_kernel(const float* __restrict__ AAt,
                                                    const float* __restrict__ BAt,
                                                    const float* __restrict__ D,
                                                    float* __restrict__ Out)
{
    extern __shared__ float lds[];
    float* Msh = lds;                 // K_ * PITCH
    float* rhs = lds + K_ * PITCH;    // K_

    const int b = blockIdx.x >> 7;
    const int i = blockIdx.x & (K_ - 1);
    const int r = threadIdx.x;

    const float* aat = AAt + (size_t)b * K_ * K_;
    for (int idx = r; idx < K_ * K_; idx += 128) {
        const int rr = idx >> 7;
        const int cc = idx & (K_ - 1);
        Msh[rr * PITCH + cc] = aat[idx];
    }
    rhs[r] = BAt[(size_t)b * K_ * K_ + (size_t)i * K_ + r];
    __syncthreads();
    Msh[r * PITCH + r] += LMBDA * D[(size_t)b * K_ * K_ + (size_t)i * K_ + r];

    // forward elimination (SPD -> no pivoting needed)
    for (int p = 0; p < K_ - 1; ++p) {
        __syncthreads();
        if (r > p) {
            const float fmul = Msh[r * PITCH + p] / Msh[p * PITCH + p];
            for (int c = p + 1; c < K_; ++c)
                Msh[r * PITCH + c] -= fmul * Msh[p * PITCH + c];
            rhs[r] -= fmul * rhs[p];
        }
    }
    // back substitution
    for (int p = K_ - 1; p >= 0; --p) {
        __syncthreads();
        if (r == p) rhs[p] /= Msh[p * PITCH + p];
        __syncthreads();
        if (r < p) rhs[r] -= Msh[r * PITCH + p] * rhs[p];
    }
    __syncthreads();
    Out[(size_t)b * K_ * K_ + (size_t)i * K_ + r] = rhs[r];
}

// ---------------------------------------------------------------------------
extern "C" void kernel_launch(void* const* d_in, const int* in_sizes, int n_in,
                              void* d_out, int out_size, void* d_ws, size_t ws_size,
                              hipStream_t stream) {
    (void)in_sizes; (void)n_in; (void)out_size; (void)ws_size;
    const float* feat_x  = (const float*)d_in[0];
    const float* feat_y  = (const float*)d_in[1];
    const float* evals_x = (const float*)d_in[2];
    const float* evals_y = (const float*)d_in[3];
    const float* etx     = (const float*)d_in[4];
    const float* ety     = (const float*)d_in[5];
    float* out = (float*)d_out;

    float* Aw  = (float*)d_ws;                       // B*K*C
    float* Bw  = Aw  + (size_t)B_ * K_ * C_;         // B*K*C
    float* AAt = Bw  + (size_t)B_ * K_ * C_;         // B*K*K
    float* BAt = AAt + (size_t)B_ * K_ * K_;         // B*K*K
    float* Dm  = BAt + (size_t)B_ * K_ * K_;         // B*K*K

    dim3 gp(C_ / 64, B_);
    proj_gemm<<<gp, 256, 0, stream>>>(etx, feat_x, Aw);
    proj_gemm<<<gp, 256, 0, stream>>>(ety, feat_y, Bw);

    dim3 gg(K_ / 64, B_);
    gram_gemm<<<gg, 256, 0, stream>>>(Aw, Aw, AAt);
    gram_gemm<<<gg, 256, 0, stream>>>(Bw, Aw, BAt);

    mask_kernel<<<B_, 128, 0, stream>>>(evals_x, evals_y, Dm);

    const size_t shmem = (size_t)(K_ * PITCH + K_) * sizeof(float);
    solve_kernel<<<B_ * K_, 128, shmem, stream>>>(AAt, BAt, Dm, out);
}